// BPCA3D_8332236554529
// MI455X (gfx1250) — compile-verified
//
#include <hip/hip_runtime.h>
#include <hip/hip_bf16.h>

typedef __attribute__((ext_vector_type(2))) float v2f;
typedef __attribute__((ext_vector_type(8))) float v8f;

#define SSTR 36864   // s-stride in floats (192*192)
#define WSTR 192     // w-stride in floats
#define HO_N 96
#define WO_N 96
#define DO_N 32
#define PLANE_STR 73728   // 2 s-slices = (b*32+dd) plane stride in floats
#define LINE_STR  384     // 2*WSTR: one wo step

// ---------------------------------------------------------------------------
// Kernel A: column sums (8) + Gram (8x8) of the N x 8 block-row matrix,
// accumulated with V_WMMA_F32_16X16X4_F32 (full fp32 matrix pipe).
// Per wmma: 8 data rows; rows 0-3 -> quadrant (M,N in 0..7), rows 4-7 ->
// quadrant (M,N in 8..15); A and B register images coincide (D += A x A).
//   ISA A layout: lane L holds A[M=L%16][K=2*(L/16)+t] in vgpr t.
//   A[m][k] = row_{(m>=8?4:0)+k}[m&7]  -> fixed per-lane offset from group base.
// Waves own contiguous group ranges; base advances incrementally (SALU only).
// ---------------------------------------------------------------------------
__global__ void bpca_stats(const float* __restrict__ x, float* __restrict__ part,
                           int numGroups, int totalWaves) {
  const int lane      = threadIdx.x & 31;
  const int waveInBlk = threadIdx.x >> 5;
  const int wid       = blockIdx.x * (blockDim.x >> 5) + waveInBlk;

  // per-lane constant offsets
  const int m    = lane & 15;
  const int half = lane >> 4;
  const int j    = m & 7;                         // column this lane owns
  const int r0   = ((m >> 3) << 2) + (half << 1); // data-row within group, vgpr0
  const int moff = (j >> 2) * SSTR + ((j >> 1) & 1) * WSTR + (j & 1);
  const int off0 = 2 * r0 + moff;                 // consecutive rows: +2 in h

  // contiguous range per wave (wave-uniform)
  const int per = (numGroups + totalWaves - 1) / totalWaves;
  int g0 = wid * per;
  int g1 = g0 + per;
  if (g1 > numGroups) g1 = numGroups;

  v8f acc = {0.f, 0.f, 0.f, 0.f, 0.f, 0.f, 0.f, 0.f};
  float ssum = 0.f;

  if (g0 < g1) {
    // decompose start once; then pure incremental updates
    int hoGrp = g0 % 12;            // 12 groups of 8 rows per ho-line
    int t1    = g0 / 12;
    int wo    = t1 % WO_N;
    int plane = t1 / WO_N;          // b*32 + dd
    int base  = plane * PLANE_STR + wo * LINE_STR + hoGrp * 16;

    for (int g = g0; g < g1; ++g) {
      float a0 = x[base + off0];
      float a1 = x[base + off0 + 2];
      v2f a; a.x = a0; a.y = a1;
      acc = __builtin_amdgcn_wmma_f32_16x16x4_f32(false, a, false, a,
                                                  (short)0, acc, false, false);
      ssum += a0 + a1;              // both values belong to column j = lane&7
      base += 16;
      if (++hoGrp == 12) {          // wave-uniform wrap handling
        hoGrp = 0; base += LINE_STR - 192;        // 12*16 already added
        if (++wo == WO_N) { wo = 0; base += PLANE_STR - WO_N * LINE_STR; }
      }
    }
  }

  // column sums: column j replicated on lanes j, j+8, j+16, j+24
  float stot = ssum;
  stot += __shfl(ssum, (lane & 7) + 8,  32);
  stot += __shfl(ssum, (lane & 7) + 16, 32);
  stot += __shfl(ssum, (lane & 7) + 24, 32);

  // Gram: quadrant0 -> lanes 0..7 (G0[v][lane] = acc[v]);
  //       quadrant3 -> lanes 24..31 (G1[v][lane-24] = acc[v])
  float g8[8];
#pragma unroll
  for (int v = 0; v < 8; ++v) {
    float q3 = __shfl(acc[v], (lane & 7) + 24, 32);
    g8[v] = acc[v] + q3;            // meaningful on lanes 0..7
  }

  __shared__ float red[8][72];
  if (lane < 8) {
    red[waveInBlk][lane] = stot;
#pragma unroll
    for (int v = 0; v < 8; ++v) red[waveInBlk][8 + v * 8 + lane] = g8[v];
  }
  __syncthreads();
  const int wavesPerBlk = blockDim.x >> 5;
  if (threadIdx.x < 72) {
    float p = 0.f;
    for (int w = 0; w < wavesPerBlk; ++w) p += red[w][threadIdx.x];
    part[blockIdx.x * 72 + threadIdx.x] = p;   // [0..7]=sums, [8..71]=G
  }
}

// ---------------------------------------------------------------------------
// Kernel B: reduce partials, form C = G - s s^T / N, 8x8 cyclic Jacobi,
// top eigenvector, deterministic sign, emit mean/pca + (v, bias) for pass 3
// ---------------------------------------------------------------------------
__global__ void bpca_solve(const float* __restrict__ part, int nParts,
                           float* __restrict__ out, long nRows,
                           float* __restrict__ vout) {
  __shared__ float red[72];
  if (threadIdx.x < 72) {
    float p = 0.f;
    for (int w = 0; w < nParts; ++w) p += part[w * 72 + threadIdx.x];
    red[threadIdx.x] = p;
  }
  __syncthreads();
  if (threadIdx.x == 0) {
    const float Nf = (float)nRows;
    float s[8], A[8][8], V[8][8];
    for (int i = 0; i < 8; ++i) s[i] = red[i];
    for (int i = 0; i < 8; ++i)
      for (int jj = 0; jj < 8; ++jj) {
        A[i][jj] = red[8 + i * 8 + jj] - s[i] * s[jj] / Nf;
        V[i][jj] = (i == jj) ? 1.f : 0.f;
      }
    for (int sweep = 0; sweep < 24; ++sweep)
      for (int p = 0; p < 7; ++p)
        for (int q = p + 1; q < 8; ++q) {
          float apq = A[p][q];
          if (fabsf(apq) < 1e-12f) continue;
          float tau = (A[q][q] - A[p][p]) / (2.f * apq);
          float t  = (tau >= 0.f ? 1.f : -1.f) / (fabsf(tau) + sqrtf(1.f + tau * tau));
          float c  = 1.f / sqrtf(1.f + t * t);
          float sn = t * c;
          for (int k = 0; k < 8; ++k) { float akp = A[k][p], akq = A[k][q];
            A[k][p] = c * akp - sn * akq; A[k][q] = sn * akp + c * akq; }
          for (int k = 0; k < 8; ++k) { float apk = A[p][k], aqk = A[q][k];
            A[p][k] = c * apk - sn * aqk; A[q][k] = sn * apk + c * aqk; }
          for (int k = 0; k < 8; ++k) { float vkp = V[k][p], vkq = V[k][q];
            V[k][p] = c * vkp - sn * vkq; V[k][q] = sn * vkp + c * vkq; }
        }
    int best = 0;
    for (int i = 1; i < 8; ++i) if (A[i][i] > A[best][best]) best = i;
    float v[8], nrm = 0.f;
    for (int i = 0; i < 8; ++i) { v[i] = V[i][best]; nrm += v[i] * v[i]; }
    nrm = 1.f / sqrtf(nrm);
    int bm = 0;
    for (int i = 1; i < 8; ++i) if (fabsf(v[i]) > fabsf(v[bm])) bm = i;
    float sgn = (v[bm] >= 0.f) ? nrm : -nrm;
    float bias = 0.f;
    for (int i = 0; i < 8; ++i) {
      v[i] *= sgn;
      float mu = s[i] / Nf;
      bias += mu * v[i];
      out[nRows + i]     = mu;     // mean
      out[nRows + 8 + i] = v[i];   // pca_components (8,1)
      vout[i] = v[i];
    }
    vout[8] = bias;
  }
}

// ---------------------------------------------------------------------------
// Kernel C: out[row] = dot(row, v) - bias.  One workgroup per ho-line
// (blockDim = 96 = 3 waves): index math once per block, threads do 4
// coalesced 8-byte loads and one coalesced store.
// ---------------------------------------------------------------------------
__global__ void bpca_project(const float* __restrict__ x,
                             const float* __restrict__ vb,
                             float* __restrict__ out) {
  const float v0 = vb[0], v1 = vb[1], v2 = vb[2], v3 = vb[3];
  const float v4 = vb[4], v5 = vb[5], v6 = vb[6], v7 = vb[7];
  const float bias = vb[8];

  const int line  = blockIdx.x;        // (b*32+dd)*96 + wo
  const int wo    = line % WO_N;
  const int plane = line / WO_N;       // b*32 + dd
  const int base  = plane * PLANE_STR + wo * LINE_STR + 2 * (int)threadIdx.x;

  float2 p0 = *(const float2*)(x + base);
  float2 p1 = *(const float2*)(x + base + WSTR);
  float2 p2 = *(const float2*)(x + base + SSTR);
  float2 p3 = *(const float2*)(x + base + SSTR + WSTR);
  out[line * HO_N + threadIdx.x] =
      p0.x * v0 + p0.y * v1 + p1.x * v2 + p1.y * v3 +
      p2.x * v4 + p2.y * v5 + p3.x * v6 + p3.y * v7 - bias;
}

extern "C" void kernel_launch(void* const* d_in, const int* in_sizes, int n_in,
                              void* d_out, int out_size, void* d_ws, size_t ws_size,
                              hipStream_t stream) {
  const float* x = (const float*)d_in[0];
  float* out = (float*)d_out;
  float* ws  = (float*)d_ws;

  const int nRows     = 8 * 32 * 96 * 96;   // 2,359,296 block-rows
  const int numGroups = nRows / 8;          // 294,912 groups of 8 rows
  const int numLines  = 8 * 32 * 96;        // 24,576 ho-lines

  int nwgA = 512;                           // stats workgroups (256 thr = 8 waves)
  while (nwgA > 1 && (size_t)(nwgA * 72 + 16) * sizeof(float) > ws_size) nwgA >>= 1;

  bpca_stats<<<nwgA, 256, 0, stream>>>(x, ws, numGroups, nwgA * 8);
  bpca_solve<<<1, 128, 0, stream>>>(ws, nwgA, out, (long)nRows, ws + nwgA * 72);
  bpca_project<<<numLines, HO_N, 0, stream>>>(x, ws + nwgA * 72, out);
}